// SeqCondAttention_41678362640993
// MI455X (gfx1250) — compile-verified
//
#include <hip/hip_runtime.h>
#include <hip/hip_bf16.h>

// ---------------- problem dims ----------------
#define BB 2
#define LL 4096
#define DD 1024
#define DI 2048          // D_INNER
#define KK 32            // key heads
#define HH 64            // head dim
#define CC 4128          // in_proj cols = 2*DI + KK
#define ND_ 28           // decay heads
#define KS_ 4            // conv taps
#define NCHUNK 64        // L / CHUNK
#define CHUNK 64

typedef __attribute__((ext_vector_type(2))) float v2f;
typedef __attribute__((ext_vector_type(8))) float v8f;

// =====================================================================
// WMMA fp32 GEMM:  C[M,N] (+)= A[M,Kd] @ B[Kd,N]   (ldb = ldc = N, lda = Kd)
// Workgroup: 256 thr (8 waves), 128x128 tile, K-chunks of 32.
// A tile: double-buffered LDS, filled by GLOBAL_LOAD_ASYNC_TO_LDS_B128
//         (ASYNCcnt pipelining: issue next chunk, s_wait_asynccnt 4).
// B tile: register-prefetched (N-guarded), stored to LDS at chunk top.
// Wave tile 32x64 -> 2x4 fragments of v_wmma_f32_16x16x4_f32.
// Requires: M % 128 == 0, Kd % 32 == 0. N guarded.
// =====================================================================
__global__ __launch_bounds__(256) void wmma_gemm_f32(
    const float* __restrict__ A, const float* __restrict__ Bm,
    float* __restrict__ Cm, int M, int N, int Kd, int addC) {
  __shared__ __align__(16) float As[2 * 128 * 36];   // [buf][m][k], stride 36
  __shared__ __align__(16) float Bs[32 * 132];       // [k][n], stride 132
  const int t    = threadIdx.x;
  const int lane = t & 31;
  const int wid  = t >> 5;
  const int wm   = (wid & 3) * 32;     // wave M offset within tile
  const int wn   = (wid >> 2) * 64;    // wave N offset within tile
  const int hsel = lane >> 4;          // 0: k0/k1, 1: k2/k3 (ISA 16x4 f32 A layout)
  const int lr   = lane & 15;
  const int m0   = blockIdx.x * 128;
  const int n0   = blockIdx.y * 128;

  // ---- async A staging setup: 4 x B128 per thread per chunk ----
  const int ar0 = t >> 3;              // 0..31
  const int ac  = (t & 7) * 4;         // float column within chunk
  const unsigned long long abase = (unsigned long long)(uintptr_t)A;
  uint32_t ldsA[4];
  int goff[4];
#pragma unroll
  for (int it = 0; it < 4; ++it) {
    const int r = it * 32 + ar0;
    ldsA[it] = (uint32_t)(uintptr_t)(&As[0]) + (uint32_t)((r * 36 + ac) * 4);
    goff[it] = (int)(((size_t)(m0 + r) * Kd + ac) * 4);   // kc = 0
  }
  auto issueA = [&](int bufsel) {
    const uint32_t lofs = bufsel ? (uint32_t)(128 * 36 * 4) : 0u;
#pragma unroll
    for (int it = 0; it < 4; ++it) {
      const uint32_t la = ldsA[it] + lofs;
      asm volatile("global_load_async_to_lds_b128 %0, %1, %2"
                   :: "v"(la), "v"(goff[it]), "s"(abase)
                   : "memory");
    }
  };

  // ---- B register prefetch (guarded) ----
  const int kr0 = t >> 5;
  const int nn4 = lane * 4;
  float4 breg[4];
  auto loadB = [&](int kc) {
#pragma unroll
    for (int it = 0; it < 4; ++it) {
      const int kr  = it * 8 + kr0;
      const int col = n0 + nn4;
      float4 v;
      if (col + 3 < N) {
        v = *(const float4*)(Bm + (size_t)(kc + kr) * N + col);
      } else {
        v.x = (col + 0 < N) ? Bm[(size_t)(kc + kr) * N + col + 0] : 0.f;
        v.y = (col + 1 < N) ? Bm[(size_t)(kc + kr) * N + col + 1] : 0.f;
        v.z = (col + 2 < N) ? Bm[(size_t)(kc + kr) * N + col + 2] : 0.f;
        v.w = (col + 3 < N) ? Bm[(size_t)(kc + kr) * N + col + 3] : 0.f;
      }
      breg[it] = v;
    }
  };
  auto storeB = [&]() {
#pragma unroll
    for (int it = 0; it < 4; ++it)
      *(float4*)(Bs + (it * 8 + kr0) * 132 + nn4) = breg[it];
  };

  v8f acc[2][4];
#pragma unroll
  for (int mt = 0; mt < 2; ++mt)
#pragma unroll
    for (int nt = 0; nt < 4; ++nt)
      acc[mt][nt] = (v8f){0.f, 0.f, 0.f, 0.f, 0.f, 0.f, 0.f, 0.f};

  // prologue: chunk 0 in flight
  issueA(0);
  loadB(0);

  const int nch = Kd / 32;
  for (int ch = 0; ch < nch; ++ch) {
    storeB();                           // B chunk ch -> LDS
    if (ch + 1 < nch) {
#pragma unroll
      for (int it = 0; it < 4; ++it) goff[it] += 128;   // advance 32 floats
      issueA((ch + 1) & 1);             // async A chunk ch+1 -> other buffer
      loadB((ch + 1) * 32);             // B chunk ch+1 -> regs
      asm volatile("s_wait_asynccnt 0x4" ::: "memory"); // chunk ch arrived
    } else {
      asm volatile("s_wait_asynccnt 0x0" ::: "memory");
    }
    __syncthreads();

    const int abuf = (ch & 1) * (128 * 36);
#pragma unroll
    for (int kk = 0; kk < 32; kk += 4) {
      const int ks = kk + 2 * hsel;
      // A frag: lanes 0-15 hold (k0,k1) of row lr; lanes 16-31 hold (k2,k3)
      v2f a0 = *(const v2f*)(As + abuf + (wm + lr) * 36 + ks);
      v2f a1 = *(const v2f*)(As + abuf + (wm + 16 + lr) * 36 + ks);
      v2f bf[4];
#pragma unroll
      for (int nt = 0; nt < 4; ++nt) {
        const int nn = wn + nt * 16 + lr;
        bf[nt].x = Bs[ks * 132 + nn];
        bf[nt].y = Bs[(ks + 1) * 132 + nn];
      }
#pragma unroll
      for (int nt = 0; nt < 4; ++nt) {
        acc[0][nt] = __builtin_amdgcn_wmma_f32_16x16x4_f32(
            false, a0, false, bf[nt], (short)0, acc[0][nt], false, false);
        acc[1][nt] = __builtin_amdgcn_wmma_f32_16x16x4_f32(
            false, a1, false, bf[nt], (short)0, acc[1][nt], false, false);
      }
    }
    __syncthreads();
  }

  // ---- epilogue: D layout VGPR i -> row (i + 8*hsel), col lr ----
#pragma unroll
  for (int mt = 0; mt < 2; ++mt) {
#pragma unroll
    for (int nt = 0; nt < 4; ++nt) {
      const int nn = n0 + wn + nt * 16 + lr;
      if (nn < N) {
#pragma unroll
        for (int i = 0; i < 8; ++i) {
          const int mm = m0 + wm + mt * 16 + hsel * 8 + i;
          const size_t ci = (size_t)mm * N + nn;
          float v = acc[mt][nt][i];
          if (addC) v += Cm[ci];
          Cm[ci] = v;
        }
      }
    }
  }
}

// =====================================================================
// Kernel 2: causal depthwise conv (on the fly) + gate/score/feature math
// grid: (L/64, B*K), block 256 (h = t&63, 4 l-lanes)
// =====================================================================
__device__ __forceinline__ float softplusf_(float x) {
  return (x > 20.f) ? x : log1pf(expf(x));
}

__global__ __launch_bounds__(256) void feature_kernel(
    const float* __restrict__ z, const float* __restrict__ conv_k,
    const float* __restrict__ conv_b, const float* __restrict__ theta,
    const float* __restrict__ decay_slopes, const float* __restrict__ anchor_slopes,
    const float* __restrict__ score_scale, const float* __restrict__ deriv_logits,
    float* __restrict__ A_re, float* __restrict__ A_im,
    float* __restrict__ P, float* __restrict__ G) {
  const int bk = blockIdx.y, b = bk >> 5, k = bk & 31;
  const int tile = blockIdx.x;
  const int t = threadIdx.x, h = t & 63, li = t >> 6;

  // softmax(deriv_logits)
  const float d0 = deriv_logits[0], d1 = deriv_logits[1], d2 = deriv_logits[2];
  const float mx = fmaxf(d0, fmaxf(d1, d2));
  const float e0 = expf(d0 - mx), e1 = expf(d1 - mx), e2 = expf(d2 - mx);
  const float wden = 1.f / (e0 + e1 + e2);
  const float w0 = e0 * wden, w1 = e1 * wden, w2 = e2 * wden;

  const int chv = k * 64 + h, chg = DI + k * 64 + h, chs = 2 * DI + k;
  const float cbv = conv_b[chv], cbg = conv_b[chg], cbs = conv_b[chs];
  float kv[KS_], kg[KS_], ksc[KS_];
#pragma unroll
  for (int j = 0; j < KS_; ++j) {
    kv[j]  = conv_k[j * CC + chv];
    kg[j]  = conv_k[j * CC + chg];
    ksc[j] = conv_k[j * CC + chs];
  }
  const float th  = theta[k * 64 + h];
  const float ssc = score_scale[k];
  const float slope = (k < ND_) ? softplusf_(decay_slopes[k])
                                : softplusf_(anchor_slopes[k - ND_]);

  for (int il = 0; il < 16; ++il) {
    const int l = tile * 64 + il * 4 + li;
    float av = cbv, ag = cbg, as = cbs;
#pragma unroll
    for (int j = 0; j < KS_; ++j) {
      const int tl = l - (KS_ - 1) + j;
      if (tl >= 0) {
        const float* zt = z + (size_t)(b * LL + tl) * CC;
        av += kv[j]  * zt[chv];
        ag += kg[j]  * zt[chg];
        as += ksc[j] * zt[chs];
      }
    }
    // gate = silu
    const float gate = ag / (1.f + expf(-ag));
    const size_t gi = (size_t)(b * LL + l) * DI + k * 64 + h;
    G[gi] = gate;

    // p_w = exp(clip(scale*s)) * time_weight
    const float sv = fminf(fmaxf(ssc * as, -20.f), 20.f);
    const float p  = expf(sv);
    const float tw = (k < ND_) ? expf(-slope * (float)(LL - 1 - l))
                               : expf(-slope * (float)l);
    const float pw = p * tw;
    if (h == 0) P[(size_t)bk * LL + l] = pw;

    const float xv  = av;
    const float phi = xv * th;
    float sphi, cphi;
    sincosf(phi, &sphi, &cphi);
    const float poly = w0 + w1 * xv - w2 * xv * xv;
    const float base = pw * poly;
    A_re[gi] = base * cphi;
    A_im[gi] = base * sphi;
  }
}

// =====================================================================
// Kernel 3a: per-chunk local inclusive scans along L (in place) + chunk totals
// grid: (NCHUNK, B*K), block 256
// =====================================================================
__global__ __launch_bounds__(256) void scan_local_kernel(
    float* __restrict__ A_re, float* __restrict__ A_im, float* __restrict__ P,
    float* __restrict__ CT_re, float* __restrict__ CT_im, float* __restrict__ CT_den) {
  __shared__ float Sre[CHUNK * 65];
  __shared__ float Sim[CHUNK * 65];
  __shared__ float Sd[CHUNK];
  const int bk = blockIdx.y, b = bk >> 5, k = bk & 31;
  const int chunk = blockIdx.x;
  const int l0 = chunk * CHUNK;
  const int t = threadIdx.x, h = t & 63, lr = t >> 6;

  for (int il = 0; il < 16; ++il) {
    const int l = il * 4 + lr;
    const size_t g = (size_t)(b * LL + l0 + l) * DI + k * 64 + h;
    Sre[l * 65 + h] = A_re[g];
    Sim[l * 65 + h] = A_im[g];
  }
  if (t < CHUNK) Sd[t] = P[(size_t)bk * LL + l0 + t];
  __syncthreads();

  if (t < 64) {
    float run = 0.f;
    for (int l = 0; l < CHUNK; ++l) { run += Sre[l * 65 + t]; Sre[l * 65 + t] = run; }
    CT_re[((size_t)bk * NCHUNK + chunk) * 64 + t] = run;
  } else if (t < 128) {
    const int hh = t - 64;
    float run = 0.f;
    for (int l = 0; l < CHUNK; ++l) { run += Sim[l * 65 + hh]; Sim[l * 65 + hh] = run; }
    CT_im[((size_t)bk * NCHUNK + chunk) * 64 + hh] = run;
  } else if (t == 128) {
    float run = 0.f;
    for (int l = 0; l < CHUNK; ++l) { run += Sd[l]; Sd[l] = run; }
    CT_den[(size_t)bk * NCHUNK + chunk] = run;
  }
  __syncthreads();

  for (int il = 0; il < 16; ++il) {
    const int l = il * 4 + lr;
    const size_t g = (size_t)(b * LL + l0 + l) * DI + k * 64 + h;
    A_re[g] = Sre[l * 65 + h];
    A_im[g] = Sim[l * 65 + h];
  }
  if (t < CHUNK) P[(size_t)bk * LL + l0 + t] = Sd[t];
}

// =====================================================================
// Kernel 3b: exclusive scan of chunk totals per channel (in place)
// =====================================================================
__global__ void scan_offsets_kernel(float* __restrict__ CT_re,
                                    float* __restrict__ CT_im,
                                    float* __restrict__ CT_den) {
  const int cid = blockIdx.x * blockDim.x + threadIdx.x;
  const int NCH = BB * KK * HH;  // 4096 channels per (re/im)
  if (cid < NCH) {
    float* p = CT_re + (size_t)(cid >> 6) * NCHUNK * 64 + (cid & 63);
    float run = 0.f;
    for (int c = 0; c < NCHUNK; ++c) { const float v = p[c * 64]; p[c * 64] = run; run += v; }
  } else if (cid < 2 * NCH) {
    const int c2 = cid - NCH;
    float* p = CT_im + (size_t)(c2 >> 6) * NCHUNK * 64 + (c2 & 63);
    float run = 0.f;
    for (int c = 0; c < NCHUNK; ++c) { const float v = p[c * 64]; p[c * 64] = run; run += v; }
  } else if (cid < 2 * NCH + BB * KK) {
    float* p = CT_den + (size_t)(cid - 2 * NCH) * NCHUNK;
    float run = 0.f;
    for (int c = 0; c < NCHUNK; ++c) { const float v = p[c]; p[c] = run; run += v; }
  }
}

// =====================================================================
// Kernel 4: apply chunk offsets, 1/den, row rsqrt-norm, norm_scale (in place)
// grid: (L/64, B*K), block 256
// =====================================================================
__global__ __launch_bounds__(256) void normalize_kernel(
    float* __restrict__ A_re, float* __restrict__ A_im,
    const float* __restrict__ P, const float* __restrict__ CT_re,
    const float* __restrict__ CT_im, const float* __restrict__ CT_den,
    const float* __restrict__ norm_scale) {
  __shared__ float wsum[8];
  const int bk = blockIdx.y, b = bk >> 5, k = bk & 31;
  const int tile = blockIdx.x;
  const int t = threadIdx.x, h = t & 63, li = t >> 6, wv = t >> 5;
  const float nsr = norm_scale[h], nsi = norm_scale[64 + h];

  for (int il = 0; il < 16; ++il) {
    const int l = tile * 64 + il * 4 + li;
    const int chunk = l >> 6;
    const float den = P[(size_t)bk * LL + l] + CT_den[bk * NCHUNK + chunk];
    const float invd = 1.f / fmaxf(den, 1e-4f);
    const size_t g  = (size_t)(b * LL + l) * DI + k * 64 + h;
    const size_t ct = ((size_t)bk * NCHUNK + chunk) * 64 + h;
    const float re = (A_re[g] + CT_re[ct]) * invd;
    const float im = (A_im[g] + CT_im[ct]) * invd;
    float sq = re * re + im * im;
#pragma unroll
    for (int o = 16; o; o >>= 1) sq += __shfl_xor(sq, o, 32);
    if ((t & 31) == 0) wsum[wv] = sq;
    __syncthreads();
    const float tot = wsum[li * 2] + wsum[li * 2 + 1];
    const float rs = rsqrtf(tot * (1.0f / 128.0f) + 1e-5f);
    A_re[g] = re * rs * nsr;
    A_im[g] = im * rs * nsi;
    __syncthreads();
  }
}

// =====================================================================
// Kernel 5: elementwise gate multiply  Y *= G  (float4)
// =====================================================================
__global__ void gate_mul_kernel(float* __restrict__ Y, const float* __restrict__ G, int n4) {
  const int i = blockIdx.x * blockDim.x + threadIdx.x;
  if (i < n4) {
    float4 y = ((float4*)Y)[i];
    const float4 g = ((const float4*)G)[i];
    y.x *= g.x; y.y *= g.y; y.z *= g.z; y.w *= g.w;
    ((float4*)Y)[i] = y;
  }
}

// =====================================================================
// host launch
// =====================================================================
extern "C" void kernel_launch(void* const* d_in, const int* in_sizes, int n_in,
                              void* d_out, int out_size, void* d_ws, size_t ws_size,
                              hipStream_t stream) {
  const float* x            = (const float*)d_in[0];
  const float* in_proj_w    = (const float*)d_in[1];
  const float* conv_k       = (const float*)d_in[2];
  const float* conv_b       = (const float*)d_in[3];
  const float* theta        = (const float*)d_in[4];
  const float* decay_slopes = (const float*)d_in[5];
  const float* anchor_slopes= (const float*)d_in[6];
  const float* score_scale  = (const float*)d_in[7];
  const float* deriv_logits = (const float*)d_in[8];
  const float* W_re         = (const float*)d_in[9];
  const float* W_im         = (const float*)d_in[10];
  const float* norm_scale   = (const float*)d_in[11];
  const float* out_proj_w   = (const float*)d_in[12];
  float* out = (float*)d_out;

  // workspace partition (floats)
  float* ws = (float*)d_ws;
  const size_t SZ_Z  = (size_t)BB * LL * CC;        // 33,816,576
  const size_t SZ_A  = (size_t)BB * LL * DI;        // 16,777,216
  const size_t SZ_P  = (size_t)BB * KK * LL;        // 262,144
  const size_t SZ_CT = (size_t)BB * KK * NCHUNK * 64;  // 262,144
  float* zbuf  = ws;                 // also reused as Yg (B*L x DI)
  float* A_re  = zbuf + SZ_Z;
  float* A_im  = A_re + SZ_A;
  float* Gbuf  = A_im + SZ_A;
  float* Pbuf  = Gbuf + SZ_A;
  float* CT_re = Pbuf + SZ_P;
  float* CT_im = CT_re + SZ_CT;
  float* CT_den= CT_im + SZ_CT;      // + B*K*NCHUNK

  const int Mrows = BB * LL;         // 8192

  // 1) z = x @ in_proj_w   (8192x1024 @ 1024x4128)
  wmma_gemm_f32<<<dim3(Mrows / 128, (CC + 127) / 128), 256, 0, stream>>>(
      x, in_proj_w, zbuf, Mrows, CC, DD, 0);

  // 2) conv + features -> A_re, A_im, P, G
  feature_kernel<<<dim3(LL / 64, BB * KK), 256, 0, stream>>>(
      zbuf, conv_k, conv_b, theta, decay_slopes, anchor_slopes,
      score_scale, deriv_logits, A_re, A_im, Pbuf, Gbuf);

  // 3) chunked scans along L
  scan_local_kernel<<<dim3(NCHUNK, BB * KK), 256, 0, stream>>>(
      A_re, A_im, Pbuf, CT_re, CT_im, CT_den);
  scan_offsets_kernel<<<dim3((2 * BB * KK * HH + BB * KK + 255) / 256), 256, 0, stream>>>(
      CT_re, CT_im, CT_den);

  // 4) normalization (in place -> Re_s, Im_s)
  normalize_kernel<<<dim3(LL / 64, BB * KK), 256, 0, stream>>>(
      A_re, A_im, Pbuf, CT_re, CT_im, CT_den, norm_scale);

  // 5) per-head mix as GEMMs: Yg = Re_s @ W_re + Im_s @ W_im (262144x64 @ 64x64)
  const int MH = BB * LL * KK;       // 262144
  wmma_gemm_f32<<<dim3(MH / 128, 1), 256, 0, stream>>>(A_re, W_re, zbuf, MH, HH, HH, 0);
  wmma_gemm_f32<<<dim3(MH / 128, 1), 256, 0, stream>>>(A_im, W_im, zbuf, MH, HH, HH, 1);

  // 6) gate
  gate_mul_kernel<<<dim3((int)(SZ_A / 4 / 256)), 256, 0, stream>>>(
      zbuf, Gbuf, (int)(SZ_A / 4));

  // 7) out = Yg @ out_proj_w   (8192x2048 @ 2048x1024)
  wmma_gemm_f32<<<dim3(Mrows / 128, DD / 128), 256, 0, stream>>>(
      zbuf, out_proj_w, out, Mrows, DD, DI, 0);
}